// GATv2_16527034155119
// MI455X (gfx1250) — compile-verified
//
#include <hip/hip_runtime.h>
#include <math.h>

// ---------------------------------------------------------------------------
// GATv2 (2 layers) for MI455X / gfx1250, fp32 end-to-end.
//  - Dense transforms: V_WMMA_F32_16X16X4_F32 (exact fp32 on the matrix pipe),
//    double-buffered LDS staging of both A and B tiles. Tile staging uses
//    GLOBAL_LOAD_ASYNC_TO_LDS_B128 (ASYNCcnt) so the copy engine runs
//    concurrently with the WMMA pipeline.
//  - Edge phase: one wave32 per edge, L2-resident gathers + f32 atomics.
// ---------------------------------------------------------------------------

typedef __attribute__((ext_vector_type(2))) float v2f;
typedef __attribute__((ext_vector_type(8))) float v8f;
typedef __attribute__((__vector_size__(4 * sizeof(int)))) int gat_v4i;

#define N_NODES 50000
#define N_EDGES 500000
#define DIN     512
#define HIDDIM  256   // H1 * DH
#define DOUT    64
#define HEADS1  4
#define NEG_SLOPE 0.2f

// ---- gfx1250 async global->LDS path (hedged: falls back to plain stores) ---
#if defined(__HIP_DEVICE_COMPILE__) && defined(__gfx1250__) && \
    __has_builtin(__builtin_amdgcn_global_load_async_to_lds_b128)
#define GAT_HAVE_ASYNC 1
#else
#define GAT_HAVE_ASYNC 0
#endif

__device__ __forceinline__ void gat_cp16(float* lds_dst, const float* gsrc) {
#if GAT_HAVE_ASYNC
    __builtin_amdgcn_global_load_async_to_lds_b128(
        (__attribute__((address_space(1))) gat_v4i*)gsrc,
        (__attribute__((address_space(3))) gat_v4i*)lds_dst, 0, 0);
#else
    *(float4*)lds_dst = *(const float4*)gsrc;
#endif
}

__device__ __forceinline__ void gat_stage_wait_barrier() {
#if GAT_HAVE_ASYNC
#if __has_builtin(__builtin_amdgcn_s_wait_asynccnt)
    __builtin_amdgcn_s_wait_asynccnt(0);
#else
    asm volatile("s_wait_asynccnt 0x0" ::: "memory");
#endif
#endif
    __syncthreads();
}

// ---------------------------------------------------------------- utilities

__global__ void gat_fill_f32(float* __restrict__ p, float v, int n) {
    int i = blockIdx.x * blockDim.x + threadIdx.x;
    if (i < n) p[i] = v;
}

__device__ inline void atomicMaxF32(float* addr, float val) {
    // Monotonic IEEE-754 trick: positives via signed-int max,
    // negatives via unsigned-int min. Safe when mixed.
    if (val >= 0.0f) {
        atomicMax((int*)addr, __float_as_int(val));
    } else {
        atomicMin((unsigned int*)addr, __float_as_uint(val));
    }
}

// ------------------------------------------------------------- WMMA GEMM
// Y[rows, Dout] = X[rows, Din] @ W[Din, Dout] + bias
// grid.x = rows/16 (rows = 50000 = 16*3125 exactly), grid.y = Dout/64.
// Block = 128 threads = 4 waves; wave w owns output cols [gy*64+w*16, +16).
// Per 64-K chunk, A (16x64) and B (64x64) are staged in double-buffered LDS;
// chunk i+1's copies are issued (async when available) before computing
// chunk i, overlapping the copy engine with 16 back-to-back WMMAs.
// fp32 WMMA layouts (ISA 7.12.2):
//   A 16x4 : VGPR0 = K0 (lanes 0-15) / K2 (lanes 16-31); VGPR1 = K1 / K3
//   B  4x16: mirrored (lane = N column, VGPR/lane-half = K row)
//   C/D    : VGPR i -> M = i + (lane>=16 ? 8 : 0), N = lane & 15
#define APITCH 68
#define ATILE  (16 * APITCH)   // 1088 floats
#define BTILE  (64 * APITCH)   // 4352 floats

__device__ __forceinline__ void gat_stage_tiles(
        const float* __restrict__ X, const float* __restrict__ W,
        float* aB, float* bB,
        int rowBase, int colGrp, int Din, int Dout, int k0, int tid) {
    // A chunk: 16 rows x 64 cols = 256 float4s, 2 per thread
#pragma unroll
    for (int j = 0; j < 2; ++j) {
        int q  = tid + j * 128;           // 0..255
        int r  = q >> 4;                  // 0..15
        int c4 = (q & 15) << 2;           // 0..60
        gat_cp16(aB + r * APITCH + c4,
                 X + (size_t)(rowBase + r) * Din + (k0 + c4));
    }
    // B chunk: 64 K-rows x 64 cols = 1024 float4s, 8 per thread
#pragma unroll
    for (int j = 0; j < 8; ++j) {
        int q  = tid + j * 128;           // 0..1023
        int kr = q >> 4;                  // 0..63
        int c4 = (q & 15) << 2;           // 0..60
        gat_cp16(bB + kr * APITCH + c4,
                 W + (size_t)(k0 + kr) * Dout + colGrp + c4);
    }
}

__global__ __launch_bounds__(128) void gat_gemm_bias_wmma(
        const float* __restrict__ X, const float* __restrict__ W,
        const float* __restrict__ bias, float* __restrict__ Y,
        int Din, int Dout) {
    __shared__ float lds[2 * ATILE + 2 * BTILE];   // ~43.5 KB

    const int tid  = threadIdx.x;
    const int wave = tid >> 5;
    const int lane = tid & 31;

    const int rowBase  = blockIdx.x * 16;
    const int colGrp   = blockIdx.y * 64;
    const int colLocal = wave * 16 + (lane & 15);
    const int arow     = lane & 15;
    const int koff     = (lane >> 4) * 2;   // 0 for lanes 0-15, 2 for 16-31

    v8f acc = {0.f, 0.f, 0.f, 0.f, 0.f, 0.f, 0.f, 0.f};

    const int nChunks = Din >> 6;
    // prologue: stage chunk 0 into buffer 0
    gat_stage_tiles(X, W, lds, lds + 2 * ATILE,
                    rowBase, colGrp, Din, Dout, 0, tid);

    for (int ci = 0; ci < nChunks; ++ci) {
        gat_stage_wait_barrier();          // buffer ci&1 ready for all waves
        if (ci + 1 < nChunks) {            // overlap next chunk's copies
            int nb = (ci + 1) & 1;
            gat_stage_tiles(X, W, lds + nb * ATILE,
                            lds + 2 * ATILE + nb * BTILE,
                            rowBase, colGrp, Din, Dout, (ci + 1) << 6, tid);
        }
        const float* aB = lds + (ci & 1) * ATILE;
        const float* bB = lds + 2 * ATILE + (ci & 1) * BTILE;
#pragma unroll
        for (int kk = 0; kk < 64; kk += 4) {
            v2f a, b;
            a.x = aB[arow * APITCH + kk + koff];
            a.y = aB[arow * APITCH + kk + koff + 1];
            b.x = bB[(kk + koff) * APITCH + colLocal];
            b.y = bB[(kk + koff + 1) * APITCH + colLocal];
            acc = __builtin_amdgcn_wmma_f32_16x16x4_f32(
                      false, a, false, b, (short)0, acc, false, false);
        }
        // no trailing barrier: next iteration's wait+barrier protects buffers
    }

    const int   col  = colGrp + colLocal;
    const float bb   = bias[col];
    const int   mrow = rowBase + ((lane >> 4) << 3);
#pragma unroll
    for (int i = 0; i < 8; ++i)
        Y[(size_t)(mrow + i) * Dout + col] = acc[i] + bb;
}

// --------------------------------------------------------- edge score pass
// e[edge,h] = sum_c leakyrelu(xl[src,h,c] + xr[dst,h,c]) * att[h,c]
// one wave32 per edge; lane covers channels {lane, lane+32} of each head.
template <int HEADS>
__global__ __launch_bounds__(256) void gat_edge_score(
        const float* __restrict__ XL, const float* __restrict__ XR,
        const int* __restrict__ srcI, const int* __restrict__ dstI,
        const float* __restrict__ att, float* __restrict__ Ebuf,
        float* __restrict__ M, int nEdges) {
    const int lane = threadIdx.x & 31;
    const int e    = blockIdx.x * 8 + (threadIdx.x >> 5);
    if (e >= nEdges) return;

    const int s = srcI[e];
    const int d = dstI[e];
    const int S = HEADS * 64;
    const float* xs = XL + (size_t)s * S;
    const float* xd = XR + (size_t)d * S;

#pragma unroll
    for (int h = 0; h < HEADS; ++h) {
        float p = 0.f;
#pragma unroll
        for (int cc = 0; cc < 2; ++cc) {
            int   c  = h * 64 + lane + cc * 32;
            float g  = xs[c] + xd[c];
            float lr = g > 0.f ? g : NEG_SLOPE * g;
            p += lr * att[h * 64 + lane + cc * 32];
        }
#pragma unroll
        for (int off = 16; off >= 1; off >>= 1)
            p += __shfl_xor(p, off, 32);
        if (lane == 0) {
            Ebuf[(size_t)e * HEADS + h] = p;
            atomicMaxF32(&M[(size_t)d * HEADS + h], p);
        }
    }
}

// ------------------------------------------------- exp + denominator pass
template <int HEADS>
__global__ void gat_edge_exp(
        float* __restrict__ Ebuf, const int* __restrict__ dstI,
        const float* __restrict__ M, float* __restrict__ Ssum, int nEdges) {
    int t = blockIdx.x * blockDim.x + threadIdx.x;
    if (t >= nEdges * HEADS) return;
    int e = t / HEADS;
    int h = t - e * HEADS;
    int d = dstI[e];
    float ex = __expf(Ebuf[t] - M[(size_t)d * HEADS + h]);
    Ebuf[t] = ex;
    atomicAdd(&Ssum[(size_t)d * HEADS + h], ex);
}

// ------------------------------------------------------- weighted scatter
template <int HEADS>
__global__ __launch_bounds__(256) void gat_edge_aggregate(
        const float* __restrict__ XL, const float* __restrict__ Ebuf,
        const float* __restrict__ Ssum,
        const int* __restrict__ srcI, const int* __restrict__ dstI,
        float* __restrict__ ACC, int nEdges) {
    const int lane = threadIdx.x & 31;
    const int e    = blockIdx.x * 8 + (threadIdx.x >> 5);
    if (e >= nEdges) return;

    const int s = srcI[e];
    const int d = dstI[e];
    const int S = HEADS * 64;

#pragma unroll
    for (int h = 0; h < HEADS; ++h) {
        float alpha = Ebuf[(size_t)e * HEADS + h] /
                      (Ssum[(size_t)d * HEADS + h] + 1e-16f);
#pragma unroll
        for (int cc = 0; cc < 2; ++cc) {
            int c = h * 64 + lane + cc * 32;
            atomicAdd(&ACC[(size_t)d * S + c], alpha * XL[(size_t)s * S + c]);
        }
    }
}

// ------------------------------------------------------ bias + ELU (layer1)
__global__ void gat_bias_elu(const float* __restrict__ ACC,
                             const float* __restrict__ bias,
                             float* __restrict__ H, int total, int mask) {
    int t = blockIdx.x * blockDim.x + threadIdx.x;
    if (t >= total) return;
    float v = ACC[t] + bias[t & mask];
    H[t] = v > 0.f ? v : expm1f(v);
}

// ------------------------------------------- bias + log_softmax (layer2 out)
__global__ __launch_bounds__(256) void gat_bias_logsoftmax(
        const float* __restrict__ ACC, const float* __restrict__ bias,
        float* __restrict__ out, int nNodes) {
    const int lane = threadIdx.x & 31;
    const int n    = blockIdx.x * 8 + (threadIdx.x >> 5);
    if (n >= nNodes) return;

    float v0 = ACC[(size_t)n * 64 + lane]      + bias[lane];
    float v1 = ACC[(size_t)n * 64 + lane + 32] + bias[lane + 32];

    float m = fmaxf(v0, v1);
#pragma unroll
    for (int off = 16; off >= 1; off >>= 1)
        m = fmaxf(m, __shfl_xor(m, off, 32));
    float s = __expf(v0 - m) + __expf(v1 - m);
#pragma unroll
    for (int off = 16; off >= 1; off >>= 1)
        s += __shfl_xor(s, off, 32);
    float lse = m + __logf(s);

    out[(size_t)n * 64 + lane]      = v0 - lse;
    out[(size_t)n * 64 + lane + 32] = v1 - lse;
}

// ---------------------------------------------------------------------------
extern "C" void kernel_launch(void* const* d_in, const int* in_sizes, int n_in,
                              void* d_out, int out_size, void* d_ws, size_t ws_size,
                              hipStream_t stream) {
    const float* x     = (const float*)d_in[0];
    const int*   ei    = (const int*)d_in[1];   // [2, E]
    const float* W1l   = (const float*)d_in[3];
    const float* b1l   = (const float*)d_in[4];
    const float* W1r   = (const float*)d_in[5];
    const float* b1r   = (const float*)d_in[6];
    const float* att1  = (const float*)d_in[7];
    const float* bias1 = (const float*)d_in[8];
    const float* W2l   = (const float*)d_in[9];
    const float* b2l   = (const float*)d_in[10];
    const float* W2r   = (const float*)d_in[11];
    const float* b2r   = (const float*)d_in[12];
    const float* att2  = (const float*)d_in[13];
    const float* bias2 = (const float*)d_in[14];
    float*       out   = (float*)d_out;

    const int* srcI = ei;
    const int* dstI = ei + N_EDGES;

    // ---- workspace carving (floats), with region reuse (~163 MB peak) ----
    float* ws   = (float*)d_ws;
    float* XL1  = ws;                                   // N*256
    float* XR1  = XL1 + (size_t)N_NODES * HIDDIM;       // N*256 (-> H after L1)
    float* ACC1 = XR1 + (size_t)N_NODES * HIDDIM;       // N*256 (-> ACC2)
    float* E1   = ACC1 + (size_t)N_NODES * HIDDIM;      // E*4   (-> E2)
    float* M1   = E1 + (size_t)N_EDGES * HEADS1;        // N*4   (-> M2)
    float* S1   = M1 + (size_t)N_NODES * HEADS1;        // N*4   (-> S2)
    // layer-2 aliases (safe: producers of the old contents are upstream)
    float* Hbuf = XR1;                                  // elu output, N*256
    float* XL2  = XL1;                                  // N*64
    float* XR2  = XL1 + (size_t)N_NODES * DOUT;         // N*64
    float* ACC2 = ACC1;                                 // N*64
    float* E2   = E1;                                   // E
    float* M2   = M1;                                   // N
    float* S2   = S1;                                   // N

    const int FB = 256;
    const dim3 gemmBlk(128);

    // ================= layer 1 =================
    gat_fill_f32<<<(N_NODES * HIDDIM + FB - 1) / FB, FB, 0, stream>>>(ACC1, 0.f, N_NODES * HIDDIM);
    gat_fill_f32<<<(N_NODES * HEADS1 + FB - 1) / FB, FB, 0, stream>>>(M1, -1e30f, N_NODES * HEADS1);
    gat_fill_f32<<<(N_NODES * HEADS1 + FB - 1) / FB, FB, 0, stream>>>(S1, 0.f, N_NODES * HEADS1);

    gat_gemm_bias_wmma<<<dim3(N_NODES / 16, HIDDIM / 64), gemmBlk, 0, stream>>>(
        x, W1l, b1l, XL1, DIN, HIDDIM);
    gat_gemm_bias_wmma<<<dim3(N_NODES / 16, HIDDIM / 64), gemmBlk, 0, stream>>>(
        x, W1r, b1r, XR1, DIN, HIDDIM);

    gat_edge_score<HEADS1><<<N_EDGES / 8, 256, 0, stream>>>(
        XL1, XR1, srcI, dstI, att1, E1, M1, N_EDGES);
    gat_edge_exp<HEADS1><<<(N_EDGES * HEADS1 + FB - 1) / FB, FB, 0, stream>>>(
        E1, dstI, M1, S1, N_EDGES);
    gat_edge_aggregate<HEADS1><<<N_EDGES / 8, 256, 0, stream>>>(
        XL1, E1, S1, srcI, dstI, ACC1, N_EDGES);

    gat_bias_elu<<<(N_NODES * HIDDIM + FB - 1) / FB, FB, 0, stream>>>(
        ACC1, bias1, Hbuf, N_NODES * HIDDIM, HIDDIM - 1);

    // ================= layer 2 =================
    gat_fill_f32<<<(N_NODES * DOUT + FB - 1) / FB, FB, 0, stream>>>(ACC2, 0.f, N_NODES * DOUT);
    gat_fill_f32<<<(N_NODES + FB - 1) / FB, FB, 0, stream>>>(M2, -1e30f, N_NODES);
    gat_fill_f32<<<(N_NODES + FB - 1) / FB, FB, 0, stream>>>(S2, 0.f, N_NODES);

    gat_gemm_bias_wmma<<<dim3(N_NODES / 16, DOUT / 64), gemmBlk, 0, stream>>>(
        Hbuf, W2l, b2l, XL2, HIDDIM, DOUT);
    gat_gemm_bias_wmma<<<dim3(N_NODES / 16, DOUT / 64), gemmBlk, 0, stream>>>(
        Hbuf, W2r, b2r, XR2, HIDDIM, DOUT);

    gat_edge_score<1><<<N_EDGES / 8, 256, 0, stream>>>(
        XL2, XR2, srcI, dstI, att2, E2, M2, N_EDGES);
    gat_edge_exp<1><<<(N_EDGES + FB - 1) / FB, FB, 0, stream>>>(
        E2, dstI, M2, S2, N_EDGES);
    gat_edge_aggregate<1><<<N_EDGES / 8, 256, 0, stream>>>(
        XL2, E2, S2, srcI, dstI, ACC2, N_EDGES);

    gat_bias_logsoftmax<<<(N_NODES + 7) / 8, 256, 0, stream>>>(
        ACC2, bias2, out, N_NODES);
}